// SoftRasterizer_18296560681660
// MI455X (gfx1250) — compile-verified
//
#include <hip/hip_runtime.h>
#include <math.h>

typedef float v2f __attribute__((ext_vector_type(2)));
typedef float v8f __attribute__((ext_vector_type(8)));

#define SEG_EPS 1e-12f
// 1/sigma * log2(e) : sigmoid computed via v_exp_f32 (exp2) directly
#define ZCOEF   14426.950408889634f
#define BIGF    3.0e38f

// One wave (32 lanes) = one tile of 16 pixels vs all faces.
// Per 16-face group, per edge k: three V_WMMA_F32_16X16X4_F32 compute
// dot(ab,ap), cross(ab,ap), |ap|^2 for the 16x16 (pixel x face) tile in
// exact fp32. Face-vertex table staged to LDS via async global->LDS copies.
__global__ __launch_bounds__(256) void soft_raster_wmma(
    const float* __restrict__ verts,   // [B,V,3] f32
    const int*   __restrict__ faces,   // [F,3]
    float*       __restrict__ out,     // [B,H,H] f32
    int B, int V, int F, int H, int Fpad)
{
  extern __shared__ float fv[];        // Fpad*8 floats (8/face, 32B-aligned rec)

  const int tid = threadIdx.x;
  const int wavesPerBlock = blockDim.x >> 5;
  const int tilesPerBatch = (H * H) >> 4;
  const int blocksPerBatch = tilesPerBatch / wavesPerBlock;
  const int b      = blockIdx.x / blocksPerBatch;
  const int blkInB = blockIdx.x % blocksPerBatch;

  // ---- Stage per-batch face-vertex xy table into LDS (CDNA5 async path) ----
  for (int e = tid; e < Fpad * 3; e += blockDim.x) {
    const int f = e / 3, k = e % 3;
    float* dst = &fv[f * 8 + k * 2];
    if (f < F) {
      const int vi = faces[f * 3 + k];
      const float* g = verts + ((size_t)b * V + (size_t)vi) * 3;
      // generic LDS pointer low 32 bits == LDS byte offset on AMDGPU
      unsigned lofs = (unsigned)(unsigned long long)(void*)dst;
      unsigned long long ga = (unsigned long long)g;
      asm volatile(
        "global_load_async_to_lds_b32 %0, %1, off\n\t"
        "global_load_async_to_lds_b32 %0, %1, off offset:4"
        :: "v"(lofs), "v"(ga) : "memory");
    } else {
      // pad face: tiny triangle far outside the image -> factor 1.0 exactly
      dst[0] = 1.0e3f + (k == 1 ? 1.0f : 0.0f);
      dst[1] = 1.0e3f + (k == 2 ? 1.0f : 0.0f);
    }
  }
  asm volatile("s_wait_asynccnt 0x0" ::: "memory");
  __syncthreads();

  // ---- Per-wave tile setup ----
  const int lane = tid & 31;
  const int half = lane >> 4;          // 0: K0/K1 & M 0..7 ; 1: K2/K3 & M 8..15
  const int m    = lane & 15;          // A-row pixel index / B-column face index
  const int wave = tid >> 5;
  const int tile = blkInB * wavesPerBlock + wave;
  const int pixBase = tile << 4;

  const int pixA  = pixBase + m;
  const float fx  = ((float)(pixA % H) + 0.5f) * (2.0f / (float)H) - 1.0f;
  const float fy  = ((float)(pixA / H) + 0.5f) * (2.0f / (float)H) - 1.0f;
  const float r2  = fx * fx + fy * fy;

  // A operand: 16x4 pixel features [x, y, x^2+y^2, 1]
  v2f Aop;
  Aop.x = half ? r2   : fx;
  Aop.y = half ? 1.0f : fy;

  float prod[8];
#pragma unroll
  for (int j = 0; j < 8; ++j) prod[j] = 1.0f;

  const int NG = Fpad >> 4;
  for (int g = 0; g < NG; ++g) {
    const int f = (g << 4) + m;        // this lane's face (column N of B)
    const float4 p01 = *(const float4*)&fv[f * 8];       // ds_load_b128
    const float2 p2  = *(const float2*)&fv[f * 8 + 4];   // ds_load_b64
    const float ax0 = p01.x, ay0 = p01.y;
    const float ax1 = p01.z, ay1 = p01.w;
    const float ax2 = p2.x,  ay2 = p2.y;

    float d2m[8], crmin[8], crmax[8];
#pragma unroll
    for (int j = 0; j < 8; ++j) { d2m[j] = BIGF; crmin[j] = BIGF; crmax[j] = -BIGF; }

#pragma unroll
    for (int k = 0; k < 3; ++k) {
      const float ax  = (k == 0) ? ax0 : (k == 1) ? ax1 : ax2;
      const float ay  = (k == 0) ? ay0 : (k == 1) ? ay1 : ay2;
      const float bx  = (k == 0) ? ax1 : (k == 1) ? ax2 : ax0;
      const float by  = (k == 0) ? ay1 : (k == 1) ? ay2 : ay0;
      const float abx = bx - ax, aby = by - ay;
      const float absq = abx * abx + aby * aby;
      const float rabs = __builtin_amdgcn_rcpf(absq + SEG_EPS);  // v_rcp_f32

      // B operands: coefficient columns over features [x, y, r2, 1]
      v2f Bd, Bc, Bq;
      Bd.x = half ? 0.0f                   : abx;            // dot(ab,ap)
      Bd.y = half ? -(abx * ax + aby * ay) : aby;
      Bc.x = half ? 0.0f                   : -aby;           // cross(ab,ap)
      Bc.y = half ? (aby * ax - abx * ay)  : abx;
      Bq.x = half ? 1.0f                   : -2.0f * ax;     // |ap|^2
      Bq.y = half ? (ax * ax + ay * ay)    : -2.0f * ay;

      v8f c0 = {};
      v8f Dd = __builtin_amdgcn_wmma_f32_16x16x4_f32(false, Aop, false, Bd,
                                                     (short)0, c0, false, false);
      v8f Dc = __builtin_amdgcn_wmma_f32_16x16x4_f32(false, Aop, false, Bc,
                                                     (short)0, c0, false, false);
      v8f Dq = __builtin_amdgcn_wmma_f32_16x16x4_f32(false, Aop, false, Bq,
                                                     (short)0, c0, false, false);

#pragma unroll
      for (int j = 0; j < 8; ++j) {      // pixel M = j + 8*half, face N = m
        const float dot = Dd[j];
        const float cr  = Dc[j];
        const float apq = Dq[j];
        float t = dot * rabs;
        t = fminf(fmaxf(t, 0.0f), 1.0f);                    // v_max clamp
        // d2 = apq - 2 t dot + t^2 absq  (fma form)
        const float d2 = fmaxf(fmaf(-t, fmaf(-t, absq, 2.0f * dot), apq), 0.0f);
        d2m[j]   = fminf(d2m[j], d2);
        crmin[j] = fminf(crmin[j], cr);
        crmax[j] = fmaxf(crmax[j], cr);
      }
    }

#pragma unroll
    for (int j = 0; j < 8; ++j) {
      // inside  <=>  all cross>=0  or  all cross<=0
      const bool inside = (crmin[j] >= 0.0f) | (crmax[j] <= 0.0f);
      const float w  = d2m[j] * ZCOEF;
      const float z2 = inside ? w : -w;                     // log2-domain arg
      // 1 - sigmoid = 1/(1 + 2^z2); TRANS rcp/exp co-execute with WMMA
      prod[j] *= __builtin_amdgcn_rcpf(1.0f + __builtin_amdgcn_exp2f(z2));
    }
  }

  // ---- product over the 16 faces per half (xor masks stay within half) ----
#pragma unroll
  for (int j = 0; j < 8; ++j) {
    float p = prod[j];
    p *= __shfl_xor(p, 1, 32);
    p *= __shfl_xor(p, 2, 32);
    p *= __shfl_xor(p, 4, 32);
    p *= __shfl_xor(p, 8, 32);
    prod[j] = p;
  }

  if (m == 0) {                         // lane 0 -> pixels 0..7, lane 16 -> 8..15
    float* o = out + (size_t)b * H * H + pixBase + 8 * half;
#pragma unroll
    for (int j = 0; j < 8; ++j) o[j] = 1.0f - prod[j];
  }
}

extern "C" void kernel_launch(void* const* d_in, const int* in_sizes, int n_in,
                              void* d_out, int out_size, void* d_ws, size_t ws_size,
                              hipStream_t stream) {
  const float* verts = (const float*)d_in[0];
  const int*   faces = (const int*)d_in[1];
  float*       out   = (float*)d_out;

  const int H = 128;                         // image_size from reference setup
  int B = out_size / (H * H); if (B < 1) B = 1;
  const int F = in_sizes[1] / 3;
  const int V = in_sizes[0] / (3 * B);
  const int Fpad = (F + 15) & ~15;

  const int threads = 256;
  const int wavesPerBlock = threads / 32;
  const int tilesPerBatch = (H * H) / 16;
  const int blocksPerBatch = tilesPerBatch / wavesPerBlock;
  dim3 grid((unsigned)(B * blocksPerBatch));
  size_t lds = (size_t)Fpad * 8 * sizeof(float);

  hipLaunchKernelGGL(soft_raster_wmma, grid, dim3(threads), lds, stream,
                     verts, faces, out, B, V, F, H, Fpad);
}